// NetInfoF_18975165514263
// MI455X (gfx1250) — compile-verified
//
#include <hip/hip_runtime.h>

typedef __attribute__((ext_vector_type(2))) float v2f;
typedef __attribute__((ext_vector_type(8))) float v8f;

#define CCOMP 5
#define HDIM 128
#define KTOT (CCOMP * HDIM)          // 640
#define EDGES_PER_WAVE 16
#define WAVES_PER_BLOCK 4
#define BLOCK (WAVES_PER_BLOCK * 32)
#define ROWPAD 132                   // 128 + 4 floats pad: conflict-free WMMA-fragment reads

__global__ __launch_bounds__(BLOCK)
void netinfof_edge_dot(const float* __restrict__ xs_c,
                       const float* __restrict__ xs_t,
                       const float* __restrict__ W,
                       const float* __restrict__ b,
                       const int*   __restrict__ esrc,
                       const int*   __restrict__ edst,
                       float* __restrict__ out,
                       int N, long long E, long long ngroups)
{
    __shared__ float w_lds[KTOT];
    __shared__ float p_lds[WAVES_PER_BLOCK][EDGES_PER_WAVE * ROWPAD];
    __shared__ float o_lds[WAVES_PER_BLOCK][EDGES_PER_WAVE];

    // Stage W (C x H = 640 floats) once per block.
    for (int i = threadIdx.x; i < KTOT; i += BLOCK) w_lds[i] = W[i];
    __syncthreads();

    const int lane = threadIdx.x & 31;
    const int wv   = threadIdx.x >> 5;
    const long long group = (long long)blockIdx.x * WAVES_PER_BLOCK + wv;
    if (group >= ngroups) return;
    const long long ebase = group * EDGES_PER_WAVE;

    const float bsum = b[0] + b[1] + b[2] + b[3] + b[4];

    v8f acc0 = {};
    v8f acc1 = {};
    v2f ones; ones[0] = 1.0f; ones[1] = 1.0f;

    const int row  = lane & 15;        // edge-in-group (A-matrix M row)
    const int koff = (lane >> 4) * 2;  // lanes 16-31 hold K+2, K+3 of same row

    for (int c = 0; c < CCOMP; ++c) {
        // Per-lane W chunk for this component (4 contiguous elements).
        const float4 wvv = *(const float4*)&w_lds[c * HDIM + lane * 4];
        const float* xc_base = xs_c + (size_t)c * (size_t)N * HDIM;
        const float* xt_base = xs_t + (size_t)c * (size_t)N * HDIM;

        // Phase 1: coalesced gather of 16 src rows and 16 dst rows (512B per
        // wave transaction), elementwise product with W, staged into LDS.
        #pragma unroll 4
        for (int i = 0; i < EDGES_PER_WAVE; ++i) {
            long long e = ebase + i;
            if (e >= E) e = E - 1;     // tail clamp (duplicate work, writes guarded)
            const int s = esrc[e];
            const int d = edst[e];
            const float4 av  = *(const float4*)(xc_base + (size_t)s * HDIM + lane * 4);
            const float4 bv4 = *(const float4*)(xt_base + (size_t)d * HDIM + lane * 4);
            float4 p;
            p.x = av.x * bv4.x * wvv.x;
            p.y = av.y * bv4.y * wvv.y;
            p.z = av.z * bv4.z * wvv.z;
            p.w = av.w * bv4.w * wvv.w;
            *(float4*)&p_lds[wv][i * ROWPAD + lane * 4] = p;
        }
        // Same-wave LDS RAW: DS ops are in-order per wave; just stop the
        // compiler from reordering across the phase boundary.
        asm volatile("" ::: "memory");

        // Phase 2: WMMA row-sum. A = 16 edges x 4 K-values (f32 fragment
        // layout: v2f = {K+koff, K+koff+1} for row `row`), B = ones, so
        // D[m][*] += sum_k A[m][k]. Two accumulators hide XDL latency.
        #pragma unroll
        for (int k0 = 0; k0 < HDIM; k0 += 8) {
            v2f a0 = *(const v2f*)&p_lds[wv][row * ROWPAD + k0 + koff];
            v2f a1 = *(const v2f*)&p_lds[wv][row * ROWPAD + k0 + 4 + koff];
            acc0 = __builtin_amdgcn_wmma_f32_16x16x4_f32(
                false, a0, false, ones, (short)0, acc0, false, false);
            acc1 = __builtin_amdgcn_wmma_f32_16x16x4_f32(
                false, a1, false, ones, (short)0, acc1, false, false);
        }
        asm volatile("" ::: "memory");
    }

    // Epilogue: D layout -> VGPR v holds row M=v (lanes 0-15) / M=v+8 (16-31),
    // all N columns identical. Lanes 0 and 16 expose the 16 row-sums via LDS.
    v8f accf = acc0 + acc1;
    #pragma unroll
    for (int v = 0; v < 8; ++v) {
        float x = accf[v];
        if (lane == 0)  o_lds[wv][v]     = x;
        if (lane == 16) o_lds[wv][8 + v] = x;
    }
    asm volatile("" ::: "memory");

    if (lane < 16) {
        long long e = ebase + lane;
        if (e < E) out[e] = o_lds[wv][lane] + bsum;
    }
}

extern "C" void kernel_launch(void* const* d_in, const int* in_sizes, int n_in,
                              void* d_out, int out_size, void* d_ws, size_t ws_size,
                              hipStream_t stream) {
    const float* xs_c = (const float*)d_in[0];   // (C,N,H) f32
    const float* xs_t = (const float*)d_in[1];   // (C,N,H) f32
    const float* W    = (const float*)d_in[2];   // (C,H)   f32
    const float* b    = (const float*)d_in[3];   // (C,)    f32
    const int*   ei   = (const int*)d_in[4];     // (2,E)   int32
    float* out = (float*)d_out;                  // (E,1)   f32

    const long long E = (long long)(in_sizes[4] / 2);
    const int N = (int)(in_sizes[0] / (CCOMP * HDIM));
    const long long ngroups = (E + EDGES_PER_WAVE - 1) / EDGES_PER_WAVE;
    const long long nblocks = (ngroups + WAVES_PER_BLOCK - 1) / WAVES_PER_BLOCK;

    netinfof_edge_dot<<<(unsigned)nblocks, BLOCK, 0, stream>>>(
        xs_c, xs_t, W, b, ei, ei + E, out, N, E, ngroups);
}